// Linear_act_sp_30992484008245
// MI455X (gfx1250) — compile-verified
//
#include <hip/hip_runtime.h>

// ---------------------------------------------------------------------------
// 2:4 sparse linear layer for MI455X (gfx1250):
//   M=8192 (B*S), N=4096 (d_out), K=4096 (d_in)
//   Pass 1: prune x 2:4 -> packed bf16 A [M, K/2] + 2-bit index words
//   Pass 2: convert weight f32 -> bf16 (33.5 MB, L2-resident)
//   Pass 3: GEMM with V_SWMMAC_F32_16X16X64_BF16 (2:4 sparse matrix cores)
// ---------------------------------------------------------------------------

typedef __attribute__((ext_vector_type(16))) __bf16 v16bf;
typedef __attribute__((ext_vector_type(32))) __bf16 v32bf;
typedef __attribute__((ext_vector_type(8)))  __bf16 v8bf;
typedef __attribute__((ext_vector_type(8)))  float  v8f;
typedef __attribute__((ext_vector_type(4)))  float  v4f;

#define M_DIM 8192
#define N_DIM 4096
#define K_DIM 4096

static __device__ __forceinline__ __bf16 f2bf(float f) {
  union { float f; unsigned u; } c; c.f = f;
  unsigned r = c.u + 0x7FFFu + ((c.u >> 16) & 1u);   // round-to-nearest-even
  unsigned short h = (unsigned short)(r >> 16);
  union { unsigned short s; __bf16 b; } o; o.s = h;
  return o.b;
}

// ---------------------------------------------------------------------------
// Kernel 1: 2:4 prune + pack. One thread handles 8 groups (32 floats) so it
// produces 16 packed bf16 values (32 B) and exactly one 32-bit index word
// (8 nibbles, nibble g = idx0 | idx1<<2, idx0 < idx1) matching the SWMMAC
// index-VGPR nibble layout.
// ---------------------------------------------------------------------------
__global__ __launch_bounds__(256) void prune_pack_kernel(
    const float* __restrict__ x, __bf16* __restrict__ apk,
    unsigned* __restrict__ idxbuf) {
  const int per_row = K_DIM / 32;                       // 128 index words/row
  const int tid = blockIdx.x * 256 + threadIdx.x;
  const int m   = tid / per_row;
  const int c32 = tid % per_row;
  const float* px = x + (size_t)m * K_DIM + (size_t)c32 * 32;

  union { __bf16 e[16]; v8bf h[2]; } outv;
  unsigned idxword = 0;

#pragma unroll
  for (int g = 0; g < 8; ++g) {
    v4f v = *(const v4f*)(px + g * 4);
    float vv[4] = { v.x, v.y, v.z, v.w };
    float av[4] = { __builtin_fabsf(v.x), __builtin_fabsf(v.y),
                    __builtin_fabsf(v.z), __builtin_fabsf(v.w) };
    int i0 = 0;
#pragma unroll
    for (int i = 1; i < 4; ++i) if (av[i] > av[i0]) i0 = i;
    int i1 = (i0 == 0) ? 1 : 0;
#pragma unroll
    for (int i = 0; i < 4; ++i) {
      if (i == i0) continue;
      if (av[i] > av[i1]) i1 = i;
    }
    const int lo = (i0 < i1) ? i0 : i1;
    const int hi = (i0 < i1) ? i1 : i0;
    outv.e[2 * g]     = f2bf(vv[lo]);
    outv.e[2 * g + 1] = f2bf(vv[hi]);
    idxword |= (unsigned)(lo | (hi << 2)) << (4 * g);
  }

  __bf16* pout = apk + (size_t)m * (K_DIM / 2) + (size_t)c32 * 16;
  *(v8bf*)pout       = outv.h[0];
  *(v8bf*)(pout + 8) = outv.h[1];
  idxbuf[(size_t)m * per_row + c32] = idxword;
}

// ---------------------------------------------------------------------------
// Kernel 2: weight f32 -> bf16 (row-major [N, K]); 8 elements per thread.
// ---------------------------------------------------------------------------
__global__ __launch_bounds__(256) void wconvert_kernel(
    const float* __restrict__ w, __bf16* __restrict__ wbf) {
  const size_t t = (size_t)blockIdx.x * 256 + threadIdx.x;
  const float* p = w + t * 8;
  v4f a = *(const v4f*)p;
  v4f b = *(const v4f*)(p + 4);
  union { __bf16 e[8]; v8bf v; } o;
  o.e[0] = f2bf(a.x); o.e[1] = f2bf(a.y); o.e[2] = f2bf(a.z); o.e[3] = f2bf(a.w);
  o.e[4] = f2bf(b.x); o.e[5] = f2bf(b.y); o.e[6] = f2bf(b.z); o.e[7] = f2bf(b.w);
  *(v8bf*)(wbf + t * 8) = o.v;
}

// ---------------------------------------------------------------------------
// Kernel 3: sparse GEMM. Block = 256 threads = 8 waves (2 M-waves x 4 N-waves).
// Wave tile = 32x64 (2x4 subtiles of 16x16). Block tile = 64(M) x 256(N).
// K loop steps 64 expanded-K per iteration (one SWMMAC).
// ---------------------------------------------------------------------------
union AFrag { v16bf v; v8bf h[2]; };
union BFrag { v32bf v; v8bf h[4]; };

__global__ __launch_bounds__(256) void gemm_swmmac_kernel(
    const __bf16* __restrict__ apk, const unsigned* __restrict__ idxbuf,
    const __bf16* __restrict__ wbf, float* __restrict__ out) {
  const int lane  = threadIdx.x & 31;
  const int wave  = threadIdx.x >> 5;
  const int waveM = wave & 1;
  const int waveN = wave >> 1;
  const int row   = lane & 15;
  const int hlf   = lane >> 4;

  const int m0 = blockIdx.y * 64 + waveM * 32;
  const int n0 = blockIdx.x * 256 + waveN * 64;

  v8f acc[2][4];
#pragma unroll
  for (int ms = 0; ms < 2; ++ms)
#pragma unroll
    for (int ns = 0; ns < 4; ++ns) acc[ms][ns] = (v8f)0.0f;

  for (int kc = 0; kc < K_DIM / 64; ++kc) {
    // ---- A fragments: stored 16x32 bf16 layout.
    // lane (hlf*16+row): V0-3 = packedK hlf*8+0..7, V4-7 = packedK 16+hlf*8+0..7
    AFrag A[2];
    int   idxw[2];
#pragma unroll
    for (int ms = 0; ms < 2; ++ms) {
      const __bf16* pa = apk + (size_t)(m0 + ms * 16 + row) * (K_DIM / 2)
                             + (size_t)kc * 32 + hlf * 8;
      A[ms].h[0] = *(const v8bf*)pa;
      A[ms].h[1] = *(const v8bf*)(pa + 16);
      idxw[ms] = idxbuf[(size_t)(m0 + ms * 16 + row) * (K_DIM / 32)
                        + (size_t)kc * 2 + hlf];
    }

    // ---- B fragments: 64x16 bf16, column-major consumption = weight rows.
    // lane (hlf*16+col): V0-7 = K hlf*16+0..15, V8-15 = K 32+hlf*16+0..15
#pragma unroll
    for (int ns = 0; ns < 4; ++ns) {
      BFrag B;
      const __bf16* pb = wbf + (size_t)(n0 + ns * 16 + row) * K_DIM
                             + (size_t)kc * 64 + hlf * 16;
      B.h[0] = *(const v8bf*)pb;
      B.h[1] = *(const v8bf*)(pb + 8);
      B.h[2] = *(const v8bf*)(pb + 32);
      B.h[3] = *(const v8bf*)(pb + 40);
#pragma unroll
      for (int ms = 0; ms < 2; ++ms) {
        acc[ms][ns] = __builtin_amdgcn_swmmac_f32_16x16x64_bf16(
            /*neg_a=*/false, A[ms].v, /*neg_b=*/false, B.v,
            acc[ms][ns], idxw[ms], /*reuse_a=*/false, /*reuse_b=*/false);
      }
    }
  }

  // ---- Epilogue: C/D layout => lane (hlf*16+n): VGPR r holds M = r + 8*hlf.
#pragma unroll
  for (int ms = 0; ms < 2; ++ms) {
#pragma unroll
    for (int ns = 0; ns < 4; ++ns) {
      const int n = n0 + ns * 16 + row;
#pragma unroll
      for (int r = 0; r < 8; ++r) {
        const int m = m0 + ms * 16 + r + 8 * hlf;
        out[(size_t)m * N_DIM + n] = acc[ms][ns][r];
      }
    }
  }
}

// ---------------------------------------------------------------------------
extern "C" void kernel_launch(void* const* d_in, const int* in_sizes, int n_in,
                              void* d_out, int out_size, void* d_ws, size_t ws_size,
                              hipStream_t stream) {
  (void)in_sizes; (void)n_in; (void)out_size; (void)ws_size;
  const float* x = (const float*)d_in[0];
  const float* w = (const float*)d_in[1];
  float* out = (float*)d_out;

  char* ws = (char*)d_ws;
  const size_t apk_bytes = (size_t)M_DIM * (K_DIM / 2) * 2;   // 33.5 MB
  const size_t idx_bytes = (size_t)M_DIM * (K_DIM / 32) * 4;  //  4.0 MB
  __bf16*   apk    = (__bf16*)ws;
  unsigned* idxbuf = (unsigned*)(ws + apk_bytes);
  __bf16*   wbf    = (__bf16*)(ws + apk_bytes + idx_bytes);   // 33.5 MB

  // Pass 1: prune+pack. (M*K/32) threads.
  {
    const int threads = (M_DIM * (K_DIM / 32));
    prune_pack_kernel<<<threads / 256, 256, 0, stream>>>(x, apk, idxbuf);
  }
  // Pass 2: weight convert. (N*K/8) threads.
  {
    const int threads = (N_DIM * K_DIM / 8);
    wconvert_kernel<<<threads / 256, 256, 0, stream>>>(w, wbf);
  }
  // Pass 3: sparse GEMM.
  {
    dim3 grid(N_DIM / 256, M_DIM / 64, 1);   // (16, 128)
    gemm_swmmac_kernel<<<grid, 256, 0, stream>>>(apk, idxbuf, wbf, out);
  }
}